// Net_996432413186
// MI455X (gfx1250) — compile-verified
//
#include <hip/hip_runtime.h>
#include <math.h>

typedef __attribute__((ext_vector_type(2))) float v2f;
typedef __attribute__((ext_vector_type(8))) float v8f;

#define N_PTS 8192
#define HD    128
#define KNN   4
#define BIGF  1e30f
#define BN_EPS 1e-5f

__device__ __forceinline__ float elu1(float x) { return x > 0.f ? x : (expf(x) - 1.f); }

// -------------------- encoder: 15 -> 128 -> 128, elu --------------------
__global__ void encode_kernel(const float* __restrict__ x_lc,
                              const float* __restrict__ W1, const float* __restrict__ b1,
                              const float* __restrict__ W2, const float* __restrict__ b2,
                              float* __restrict__ out) {
  __shared__ float xin[15];
  __shared__ float h1[HD];
  int n = blockIdx.x;
  int c = threadIdx.x;                 // 128 threads
  if (c < 15) xin[c] = x_lc[n * 15 + c];
  __syncthreads();
  float a = b1[c];
#pragma unroll
  for (int i = 0; i < 15; i++) a += xin[i] * W1[i * HD + c];
  h1[c] = elu1(a);
  __syncthreads();
  float a2 = b2[c];
  for (int i = 0; i < HD; i++) a2 += h1[i] * W2[i * HD + c];
  out[n * HD + c] = elu1(a2);
}

// -------------------- per-point squared norms --------------------
__global__ void rownorm_kernel(const float* __restrict__ f, float* __restrict__ sq) {
  int n = blockIdx.x * blockDim.x + threadIdx.x;
  if (n >= N_PTS) return;
  const float* r = f + (size_t)n * HD;
  float s = 0.f;
  for (int i = 0; i < HD; i++) { float v = r[i]; s += v * v; }
  sq[n] = s;
}

// -------------------- fused WMMA Gram + top-4 KNN --------------------
// 64 blocks x 256 threads (8 waves). Wave w owns row tile rt = blockIdx*8+w
// (16 rows). Streams all 512 column tiles; Gram tile via v_wmma_f32_16x16x4_f32.
// A-fragments (K=128 -> 32 x v2f) are kept register-resident via full unroll.
__global__ void __launch_bounds__(256) knn_kernel(const float* __restrict__ feat,
                                                  const float* __restrict__ sq,
                                                  const int* __restrict__ batch,
                                                  int* __restrict__ idx_out) {
  __shared__ float Ylds[16 * HD];          // column tile features (row-major)
  __shared__ float sqY[16];
  __shared__ int   batY[16];
  __shared__ float tile[8][16][16];        // per-wave 16x16 gram tile [M][N]
  const int lane = threadIdx.x & 31;
  const int w    = threadIdx.x >> 5;
  const int rt   = blockIdx.x * 8 + w;
  const int hl   = lane & 15;
  const int hsel = lane >> 4;              // 0: K=k0,k0+1  1: K=k0+2,k0+3
  const int Mrow = rt * 16 + hl;

  // preload A fragments: this wave's 16 rows, K=128 -> 32 steps of 4.
  // FULL unroll => constant indices => stays in VGPRs (no scratch spill).
  v2f afrag[32];
  {
    const float* rp = feat + (size_t)Mrow * HD + 2 * hsel;
#pragma unroll
    for (int s = 0; s < 32; s++) { afrag[s].x = rp[4 * s]; afrag[s].y = rp[4 * s + 1]; }
  }

  // top-4 (ascending) state; lanes 0..15 own one row each in the scan phase
  float t0 = BIGF, t1 = BIGF, t2 = BIGF, t3 = BIGF;
  const int Rrow = rt * 16 + lane;
  int i0 = Mrow, i1 = Mrow, i2 = Mrow, i3 = Mrow;
  const int   bR  = (lane < 16) ? batch[Rrow] : 0;
  const float sqR = (lane < 16) ? sq[Rrow] : 0.f;

  for (int ct = 0; ct < N_PTS / 16; ct++) {
    const int C0 = ct * 16;
    for (int e = threadIdx.x; e < 16 * HD; e += 256)
      Ylds[e] = feat[(size_t)C0 * HD + e];
    if (threadIdx.x < 16) {
      sqY[threadIdx.x]  = sq[C0 + threadIdx.x];
      batY[threadIdx.x] = batch[C0 + threadIdx.x];
    }
    __syncthreads();

    // Gram tile: D = X(16xK) * Y^T(Kx16), K=128 (fully unrolled)
    v8f acc = {};
#pragma unroll
    for (int s = 0; s < 32; s++) {
      const float* bp = &Ylds[hl * HD + 4 * s + 2 * hsel];
      v2f b; b.x = bp[0]; b.y = bp[1];
      acc = __builtin_amdgcn_wmma_f32_16x16x4_f32(false, afrag[s], false, b,
                                                  (short)0, acc, false, false);
    }
#pragma unroll
    for (int g = 0; g < 8; g++) tile[w][hsel * 8 + g][hl] = acc[g];
    __syncthreads();

    if (lane < 16) {
#pragma unroll 4
      for (int c = 0; c < 16; c++) {
        int C = C0 + c;
        float d = sqR + sqY[c] - 2.f * tile[w][lane][c];
        bool valid = (batY[c] == bR) && (C != Rrow);
        if (valid && d < t3) {
          if (d < t2) {
            t3 = t2; i3 = i2;
            if (d < t1) {
              t2 = t1; i2 = i1;
              if (d < t0) { t1 = t0; i1 = i0; t0 = d; i0 = C; }
              else        { t1 = d;  i1 = C; }
            } else { t2 = d; i2 = C; }
          } else { t3 = d; i3 = C; }
        }
      }
    }
    __syncthreads();
  }
  if (lane < 16) {
    idx_out[Rrow * 4 + 0] = i0; idx_out[Rrow * 4 + 1] = i1;
    idx_out[Rrow * 4 + 2] = i2; idx_out[Rrow * 4 + 3] = i3;
  }
}

// -------------------- BN stat init --------------------
__global__ void bninit_kernel(float* bnsum, float* bnss) {
  int t = threadIdx.x;
  if (t < HD) { bnsum[t] = 0.f; bnss[t] = 0.f; }
}

// -------------------- EdgeConv GEMM: h = elu(E @ W + b), E = [xi, xj-xi] --------------------
// 64 blocks x 256 threads; each block handles 32 row-tiles of 16 edge-rows,
// all 128 output cols (8 waves x 16 cols), K=256 in 4 LDS chunks of 64.
__global__ void __launch_bounds__(256) edgeconv_kernel(const float* __restrict__ feat,
                                                       const int* __restrict__ idx,
                                                       const float* __restrict__ W,
                                                       const float* __restrict__ bias,
                                                       float* __restrict__ hout,
                                                       float* __restrict__ bnsum,
                                                       float* __restrict__ bnss) {
  __shared__ float E[16 * 256];    // 16 rows x 256 features (16KB)
  __shared__ float Wt[128 * 64];   // transposed weight chunk: Wt[c*64+kl] (32KB)
  const int lane = threadIdx.x & 31;
  const int w    = threadIdx.x >> 5;
  const int hl   = lane & 15;
  const int hsel = lane >> 4;
  const int col  = w * 16 + hl;    // this lane's output column
  const float bcol = bias[col];
  float s_sum = 0.f, s_ss = 0.f;

  for (int it = 0; it < 32; it++) {
    const int rt = blockIdx.x * 32 + it;        // 2048 row tiles total
    __syncthreads();
    {   // build E tile: flat edge row R = n*4 + kk
      int r = threadIdx.x >> 4;                 // 0..15
      int cseg = (threadIdx.x & 15) * 16;
      int R = rt * 16 + r;
      int n = R >> 2, kk = R & 3;
      int j = idx[n * 4 + kk];
      const float* xn = feat + (size_t)n * HD;
      const float* xj = feat + (size_t)j * HD;
#pragma unroll
      for (int q = 0; q < 16; q++) {
        int c = cseg + q;
        E[r * 256 + c] = (c < HD) ? xn[c] : (xj[c - HD] - xn[c - HD]);
      }
    }
    v8f acc = {};
    for (int kc = 0; kc < 4; kc++) {
      __syncthreads();
      {   // load transposed W chunk: Wt[c][kl] = W[(kc*64+kl)*128 + c]
        int cc  = threadIdx.x >> 1;
        int kl0 = (threadIdx.x & 1) * 32;
#pragma unroll 8
        for (int q = 0; q < 32; q++)
          Wt[cc * 64 + kl0 + q] = W[(size_t)(kc * 64 + kl0 + q) * HD + cc];
      }
      __syncthreads();
#pragma unroll
      for (int s = 0; s < 16; s++) {
        int kl = 4 * s + 2 * hsel;
        v2f a; a.x = E[hl * 256 + kc * 64 + kl]; a.y = E[hl * 256 + kc * 64 + kl + 1];
        v2f b; b.x = Wt[col * 64 + kl];          b.y = Wt[col * 64 + kl + 1];
        acc = __builtin_amdgcn_wmma_f32_16x16x4_f32(false, a, false, b,
                                                    (short)0, acc, false, false);
      }
    }
    // epilogue: bias + elu + store + BN stats
#pragma unroll
    for (int g = 0; g < 8; g++) {
      int M = hsel * 8 + g;
      int R = rt * 16 + M;
      float v = elu1(acc[g] + bcol);
      hout[(size_t)R * HD + col] = v;
      s_sum += v; s_ss += v * v;
    }
  }
  atomicAdd(&bnsum[col], s_sum);
  atomicAdd(&bnss[col],  s_ss);
}

// -------------------- BN finalize --------------------
__global__ void bnfin_kernel(const float* bnsum, const float* bnss,
                             const float* gamma, const float* beta,
                             float* scale, float* shift) {
  int c = threadIdx.x;
  const float inv = 1.f / (float)(N_PTS * KNN);
  float m = bnsum[c] * inv;
  float v = bnss[c] * inv - m * m;
  float sc = gamma[c] * rsqrtf(v + BN_EPS);
  scale[c] = sc;
  shift[c] = beta[c] - m * sc;
}

// -------------------- BN apply + max over k + residual --------------------
__global__ void bnmaxres_kernel(const float* __restrict__ h,
                                const float* __restrict__ scale,
                                const float* __restrict__ shift,
                                const float* __restrict__ res,
                                float* __restrict__ out) {
  int t = blockIdx.x * blockDim.x + threadIdx.x;    // n*HD + c
  if (t >= N_PTS * HD) return;
  int n = t >> 7, c = t & 127;
  float sc = scale[c], sh = shift[c];
  float m = -BIGF;
#pragma unroll
  for (int kk = 0; kk < KNN; kk++)
    m = fmaxf(m, h[(size_t)(n * KNN + kk) * HD + c] * sc + sh);
  out[t] = (res ? res[t] : 0.f) + m;
}

// -------------------- head MLP: 128 -> 32 -> 16 -> 8 --------------------
__global__ void head_kernel(const float* __restrict__ f3,
                            const float* __restrict__ Wo1, const float* __restrict__ bo1,
                            const float* __restrict__ Wo2, const float* __restrict__ bo2,
                            const float* __restrict__ Wo3, const float* __restrict__ bo3,
                            float* __restrict__ out) {
  __shared__ float xr[HD];
  __shared__ float o1[32];
  __shared__ float o2[16];
  int n = blockIdx.x, t = threadIdx.x;   // 32 threads
  for (int i = t; i < HD; i += 32) xr[i] = f3[(size_t)n * HD + i];
  __syncthreads();
  float a = bo1[t];
  for (int i = 0; i < HD; i++) a += xr[i] * Wo1[i * 32 + t];
  o1[t] = elu1(a);
  __syncthreads();
  if (t < 16) {
    float a2 = bo2[t];
    for (int i = 0; i < 32; i++) a2 += o1[i] * Wo2[i * 16 + t];
    o2[t] = elu1(a2);
  }
  __syncthreads();
  if (t < 8) {
    float a3 = bo3[t];
    for (int i = 0; i < 16; i++) a3 += o2[i] * Wo3[i * 8 + t];
    out[n * 8 + t] = a3;
  }
}

extern "C" void kernel_launch(void* const* d_in, const int* in_sizes, int n_in,
                              void* d_out, int out_size, void* d_ws, size_t ws_size,
                              hipStream_t stream) {
  (void)in_sizes; (void)n_in; (void)out_size; (void)ws_size;
  const float* x_lc  = (const float*)d_in[0];
  const int*   batch = (const int*)d_in[1];
  const float* We1 = (const float*)d_in[2];  const float* be1 = (const float*)d_in[3];
  const float* We2 = (const float*)d_in[4];  const float* be2 = (const float*)d_in[5];
  const float* Wc[3] = {(const float*)d_in[6], (const float*)d_in[10], (const float*)d_in[14]};
  const float* bc[3] = {(const float*)d_in[7], (const float*)d_in[11], (const float*)d_in[15]};
  const float* gg[3] = {(const float*)d_in[8], (const float*)d_in[12], (const float*)d_in[16]};
  const float* bt[3] = {(const float*)d_in[9], (const float*)d_in[13], (const float*)d_in[17]};
  const float* Wo1 = (const float*)d_in[18]; const float* bo1 = (const float*)d_in[19];
  const float* Wo2 = (const float*)d_in[20]; const float* bo2 = (const float*)d_in[21];
  const float* Wo3 = (const float*)d_in[22]; const float* bo3 = (const float*)d_in[23];

  float* ws = (float*)d_ws;
  const size_t NH = (size_t)N_PTS * HD;
  float* f[4] = { ws, ws + NH, ws + 2 * NH, ws + 3 * NH };
  float* sq   = ws + 4 * NH;
  int*   idxb = (int*)(ws + 4 * NH + N_PTS);
  float* hbuf = ws + 4 * NH + N_PTS + (size_t)N_PTS * KNN;
  float* bnsum = hbuf + (size_t)N_PTS * KNN * HD;
  float* bnss  = bnsum + HD;
  float* scale = bnss + HD;
  float* shift = scale + HD;

  encode_kernel<<<N_PTS, HD, 0, stream>>>(x_lc, We1, be1, We2, be2, f[0]);
  for (int l = 0; l < 3; l++) {
    rownorm_kernel<<<N_PTS / 256, 256, 0, stream>>>(f[l], sq);
    knn_kernel<<<64, 256, 0, stream>>>(f[l], sq, batch, idxb);
    bninit_kernel<<<1, HD, 0, stream>>>(bnsum, bnss);
    edgeconv_kernel<<<64, 256, 0, stream>>>(f[l], idxb, Wc[l], bc[l], hbuf, bnsum, bnss);
    bnfin_kernel<<<1, HD, 0, stream>>>(bnsum, bnss, gg[l], bt[l], scale, shift);
    bnmaxres_kernel<<<(N_PTS * HD) / 256, 256, 0, stream>>>(hbuf, scale, shift,
                                                            (l == 0) ? nullptr : f[l], f[l + 1]);
  }
  head_kernel<<<N_PTS, 32, 0, stream>>>(f[3], Wo1, bo1, Wo2, bo2, Wo3, bo3, (float*)d_out);
}